// GNNSurrogate_57440892617447
// MI455X (gfx1250) — compile-verified
//
#include <hip/hip_runtime.h>

// GNN surrogate for MI455X (gfx1250, wave32, WMMA f32).
// Compute-bound (~78 GFLOP vs ~25 MB HBM). adj A-fragments cached in 64 VGPRs
// (read from L2 exactly once per workgroup); local^T staged in LDS so adj-GEMM
// B-fragments are single aligned b64 LDS loads.

typedef __attribute__((ext_vector_type(2))) float v2f;
typedef __attribute__((ext_vector_type(8))) float v8f;

#define NQ      128   // nodes per graph
#define INFEAT  3
#define HID     32
#define NLAYERS 3
#define PITCH   36    // pitch (floats) for row-major node-feature buffers
#define PT      132   // pitch (floats) for transposed local buffer [HID][NQ]

#define WMMA_F32(A, B, C) \
    __builtin_amdgcn_wmma_f32_16x16x4_f32(false, (A), false, (B), (short)0, (C), false, false)

__launch_bounds__(256)
__global__ void gnn_surrogate_kernel(const float* __restrict__ x,     // [B,NQ,3]
                                     const float* __restrict__ adj,   // [NQ,NQ]
                                     const float* __restrict__ Wl,    // [3,HID]
                                     const float* __restrict__ bl,    // [HID]
                                     const float* __restrict__ W1,    // [L,HID,HID]
                                     const float* __restrict__ b1,    // [L,HID]
                                     const float* __restrict__ W2,    // [L,HID,HID]
                                     const float* __restrict__ b2,    // [L,HID]
                                     const float* __restrict__ Wro,   // [HID,1]
                                     const float* __restrict__ bro,   // [1]
                                     float* __restrict__ out)         // [B,1]
{
    __shared__ float bufH[NQ * PITCH];   // h, row-major [node][hid]      (A of GEMM1)
    __shared__ float bufR[NQ * PITCH];   // relu(t), row-major            (A of GEMM2) + readout scratch
    __shared__ float bufL[HID * PT];     // local^T [hid][node]           (B of adj GEMM)
    __shared__ float red[32];

    const int tid  = threadIdx.x;
    const int b    = blockIdx.x;
    const int lane = tid & 31;
    const int wave = tid >> 5;           // 8 waves; wave w owns node rows [16w, 16w+16)
    const int lo   = lane & 15;
    const int hi   = lane >> 4;
    const int m0   = wave << 4;

    // ---- cache this wave's adj rows as WMMA A-fragments: 32 x v2f = 64 VGPRs ----
    // afrag[k] lane L = adj[m0 + L%16][4k + 2*(L/16) .. +1]   (ISA 16x4 f32 A layout)
    v2f afrag[NQ / 4];
#pragma unroll
    for (int k = 0; k < NQ / 4; ++k)
        afrag[k] = *(const v2f*)(adj + (size_t)(m0 + lo) * NQ + 4 * k + 2 * hi);

    // ---- lift: h = x @ W_lift + b_lift (K=3, VALU, negligible) ----
    {
        const int n  = tid >> 1;
        const int c0 = (tid & 1) << 4;
        const float* xp = x + ((size_t)b * NQ + n) * INFEAT;
        const float x0 = xp[0], x1 = xp[1], x2 = xp[2];
#pragma unroll
        for (int c = 0; c < 16; ++c) {
            const int col = c0 + c;
            bufH[n * PITCH + col] =
                x0 * Wl[0 * HID + col] + x1 * Wl[1 * HID + col] + x2 * Wl[2 * HID + col] + bl[col];
        }
    }
    __syncthreads();

    for (int l = 0; l < NLAYERS; ++l) {
        const float* W1l = W1 + l * HID * HID;
        const float* b1l = b1 + l * HID;
        const float* W2l = W2 + l * HID * HID;
        const float* b2l = b2 + l * HID;

        // ---- t = relu(h @ W1 + b1): bufH -> bufR (wave-local rows, no barrier) ----
        {
            v8f acc0 = {}, acc1 = {};
#pragma unroll
            for (int k = 0; k < HID; k += 4) {
                const int ka = k + 2 * hi;
                v2f a = *(const v2f*)&bufH[(m0 + lo) * PITCH + ka];   // one A feeds both n-tiles
                v2f b0, b1f;
                b0.x  = W1l[ka * HID + lo];        b0.y  = W1l[(ka + 1) * HID + lo];
                b1f.x = W1l[ka * HID + 16 + lo];   b1f.y = W1l[(ka + 1) * HID + 16 + lo];
                acc0 = WMMA_F32(a, b0, acc0);
                acc1 = WMMA_F32(a, b1f, acc1);
            }
            const float bias0 = b1l[lo], bias1 = b1l[16 + lo];
#pragma unroll
            for (int r = 0; r < 8; ++r) {
                const int row = (m0 + r + 8 * hi) * PITCH;
                const float v0 = acc0[r] + bias0, v1 = acc1[r] + bias1;
                bufR[row + lo]      = v0 > 0.0f ? v0 : 0.0f;
                bufR[row + 16 + lo] = v1 > 0.0f ? v1 : 0.0f;
            }
        }

        // ---- local = t @ W2 + b2: bufR -> bufL (stored TRANSPOSED [hid][node]) ----
        {
            v8f acc0 = {}, acc1 = {};
#pragma unroll
            for (int k = 0; k < HID; k += 4) {
                const int ka = k + 2 * hi;
                v2f a = *(const v2f*)&bufR[(m0 + lo) * PITCH + ka];
                v2f b0, b1f;
                b0.x  = W2l[ka * HID + lo];        b0.y  = W2l[(ka + 1) * HID + lo];
                b1f.x = W2l[ka * HID + 16 + lo];   b1f.y = W2l[(ka + 1) * HID + 16 + lo];
                acc0 = WMMA_F32(a, b0, acc0);
                acc1 = WMMA_F32(a, b1f, acc1);
            }
            const float bias0 = b2l[lo], bias1 = b2l[16 + lo];
#pragma unroll
            for (int r = 0; r < 8; ++r) {
                const int node = m0 + r + 8 * hi;
                bufL[(lo)      * PT + node] = acc0[r] + bias0;   // local^T[hid][node]
                bufL[(16 + lo) * PT + node] = acc1[r] + bias1;
            }
        }
        __syncthreads();   // adj GEMM reads ALL of bufL

        // ---- h = adj @ local: A from cached VGPR fragments, B = aligned b64 LDS ----
        {
            v8f h0 = {}, h1 = {};
#pragma unroll
            for (int k = 0; k < NQ / 4; ++k) {
                const int ka = 4 * k + 2 * hi;
                v2f b0  = *(const v2f*)&bufL[(lo)      * PT + ka];   // local[ka..ka+1][lo]
                v2f b1f = *(const v2f*)&bufL[(16 + lo) * PT + ka];
                h0 = WMMA_F32(afrag[k], b0, h0);
                h1 = WMMA_F32(afrag[k], b1f, h1);
            }
            __syncthreads();   // all bufL reads done -> next layer may overwrite it
#pragma unroll
            for (int r = 0; r < 8; ++r) {
                const int row = (m0 + r + 8 * hi) * PITCH;
                bufH[row + lo]      = h0[r];
                bufH[row + 16 + lo] = h1[r];
            }
        }
        __syncthreads();   // h complete (needed before final readout)
    }

    // ---- readout: out[b] = (1/NQ) * sum_n h[n,:] . W_ro + b_ro ----
    {
        const int col = tid & 31;
        const int rg  = tid >> 5;
        float s = 0.0f;
#pragma unroll
        for (int r = 0; r < 16; ++r) s += bufH[(rg * 16 + r) * PITCH + col];
        bufR[tid] = s;
    }
    __syncthreads();
    if (tid < 32) {
        float s = 0.0f;
#pragma unroll
        for (int g = 0; g < 8; ++g) s += bufR[g * 32 + tid];
        red[tid] = s * Wro[tid];
    }
    __syncthreads();
    if (tid == 0) {
        float s = 0.0f;
#pragma unroll
        for (int c = 0; c < 32; ++c) s += red[c];
        out[b] = s * (1.0f / (float)NQ) + bro[0];
    }
}

extern "C" void kernel_launch(void* const* d_in, const int* in_sizes, int n_in,
                              void* d_out, int out_size, void* d_ws, size_t ws_size,
                              hipStream_t stream) {
    const float* x   = (const float*)d_in[0];
    const float* adj = (const float*)d_in[1];
    const float* Wl  = (const float*)d_in[2];
    const float* bl  = (const float*)d_in[3];
    const float* W1  = (const float*)d_in[4];
    const float* b1  = (const float*)d_in[5];
    const float* W2  = (const float*)d_in[6];
    const float* b2  = (const float*)d_in[7];
    const float* Wro = (const float*)d_in[8];
    const float* bro = (const float*)d_in[9];
    float* out = (float*)d_out;

    const int batch = in_sizes[0] / (NQ * INFEAT);   // 16384
    gnn_surrogate_kernel<<<batch, 256, 0, stream>>>(x, adj, Wl, bl, W1, b1, W2, b2, Wro, bro, out);
}